// DescriptorLoss_43619687858207
// MI455X (gfx1250) — compile-verified
//
#include <hip/hip_runtime.h>

typedef __attribute__((ext_vector_type(2))) float v2f;
typedef __attribute__((ext_vector_type(8))) float v8f;

namespace {
constexpr int kB    = 4;
constexpr int kHc   = 60;
constexpr int kWc   = 60;
constexpr int kD    = 64;
constexpr int kM    = kHc * kWc;   // 3600
constexpr int kMT   = kM / 16;     // 225 16-wide tiles
constexpr int kG    = 8;
constexpr int kFull = kHc * kG;    // 480
constexpr float kR2 = 56.25f;      // (8 - 0.5)^2
}

// ---------------------------------------------------------------------------
// Kernel 1: warp cell centers through homographies + per-cell 8x8 mask product
// + epilogue table ep[b,n] = {ccy(n), ccx(n), vm(b,n), 0}.
// ---------------------------------------------------------------------------
__global__ void prep_kernel(const float* __restrict__ homog,
                            const float* __restrict__ vmask,
                            float2* __restrict__ wcc,
                            float4* __restrict__ ep,
                            float* __restrict__ sum_vm) {
  int idx = blockIdx.x * blockDim.x + threadIdx.x;
  if (idx >= kB * kM) return;
  int b = idx / kM;
  int m = idx - b * kM;
  int i = m / kWc;
  int j = m - i * kWc;
  float cx = j * (float)kG + (float)(kG / 2);
  float cy = i * (float)kG + (float)(kG / 2);
  const float* H = homog + b * 9;
  float w0 = H[0] * cx + H[1] * cy + H[2];
  float w1 = H[3] * cx + H[4] * cy + H[5];
  float w2 = H[6] * cx + H[7] * cy + H[8];
  float inv = 1.0f / w2;
  wcc[idx] = make_float2(w1 * inv, w0 * inv);  // warped (y, x)

  const float* base = vmask + ((size_t)b * kFull + (size_t)i * kG) * kFull
                            + (size_t)j * kG;
  float prod = 1.0f;
  for (int yy = 0; yy < kG; ++yy) {
    const float* row = base + (size_t)yy * kFull;
#pragma unroll
    for (int xx = 0; xx < kG; ++xx) prod *= row[xx];
  }
  ep[idx] = make_float4(cy, cx, prod, 0.0f);
  atomicAdd(sum_vm, prod);
}

// ---------------------------------------------------------------------------
// Kernel 2: fused batched WMMA GEMM (3600x3600x64 f32) + loss epilogue + sum.
// Each wave owns one 16-row M-tile (A tile in 32 VGPRs) and streams PAIRS of
// N-tiles with two interleaved, independent WMMA accumulator chains.
// ---------------------------------------------------------------------------
__global__ void __launch_bounds__(128, 1)
gemm_loss_kernel(const float* __restrict__ desc,
                 const float* __restrict__ wdesc,
                 const float2* __restrict__ wcc,
                 const float4* __restrict__ ep,
                 double* __restrict__ acc_out) {
  const int lane = threadIdx.x & 31;
  const int wave = threadIdx.x >> 5;
  const int m_tile = blockIdx.x * 4 + wave;
  if (m_tile >= kMT) return;               // uniform per wave: EXEC stays all-1s
  const int b = blockIdx.z;
  const int half = lane >> 4;              // K sub-pair select (ISA A layout)
  const int lrow = lane & 15;              // M row (A) / N col (B)

  // A fragments: lane holds row (m_tile*16 + lrow), K pairs {4k+2h, 4k+2h+1}
  const float* Abase =
      desc + (((size_t)b * kM) + (size_t)m_tile * 16 + lrow) * kD + 2 * half;
  v2f afrag[16];
#pragma unroll
  for (int kk = 0; kk < 16; ++kk)
    afrag[kk] = *(const v2f*)(Abase + kk * 4);

  // Warped centers for this wave's 8 output rows per lane (C/D layout:
  // VGPR r -> M = r + 8*half).
  float wyv[8], wxv[8];
#pragma unroll
  for (int r = 0; r < 8; ++r) {
    int mg = m_tile * 16 + r + 8 * half;
    float2 p = wcc[b * kM + mg];
    wyv[r] = p.x;
    wxv[r] = p.y;
  }

  double partial = 0.0;

  // Epilogue for one 16x16 tile held in c (this lane's 8 dots, column n).
  auto epilogue = [&](int nt_, const v8f& c) {
    int n = nt_ * 16 + lrow;
    float4 e = ep[b * kM + n];  // {ccy, ccx, vm, 0}
    float lsum = 0.f;
#pragma unroll
    for (int r = 0; r < 8; ++r) {
      float dy = e.x - wyv[r];
      float dx = e.y - wxv[r];
      float dot = c[r];
      float pos = fmaxf(0.f, 1.0f - dot);
      float neg = fmaxf(0.f, dot - 0.2f);
      bool s = (dy * dy + dx * dx) <= kR2;
      lsum += s ? (250.0f * pos) : neg;
    }
    partial += (double)(e.z * lsum);
  };

  const int stride = (int)gridDim.y * 2;
  for (int nt = (int)blockIdx.y * 2; nt < kMT; nt += stride) {
    const float* B0 =
        wdesc + (((size_t)b * kM) + (size_t)nt * 16 + lrow) * kD + 2 * half;
    if (nt + 1 < kMT) {  // uniform branch: two tiles, two independent chains
      const float* B1 = B0 + (size_t)16 * kD;
      v8f c0 = {0.f, 0.f, 0.f, 0.f, 0.f, 0.f, 0.f, 0.f};
      v8f c1 = {0.f, 0.f, 0.f, 0.f, 0.f, 0.f, 0.f, 0.f};
#pragma unroll
      for (int kk = 0; kk < 16; ++kk) {
        v2f b0 = *(const v2f*)(B0 + kk * 4);
        v2f b1 = *(const v2f*)(B1 + kk * 4);
        c0 = __builtin_amdgcn_wmma_f32_16x16x4_f32(false, afrag[kk], false, b0,
                                                   (short)0, c0, false, false);
        c1 = __builtin_amdgcn_wmma_f32_16x16x4_f32(false, afrag[kk], false, b1,
                                                   (short)0, c1, false, false);
      }
      epilogue(nt, c0);
      epilogue(nt + 1, c1);
    } else {  // odd tail tile (kMT = 225)
      v8f c0 = {0.f, 0.f, 0.f, 0.f, 0.f, 0.f, 0.f, 0.f};
#pragma unroll
      for (int kk = 0; kk < 16; ++kk) {
        v2f b0 = *(const v2f*)(B0 + kk * 4);
        c0 = __builtin_amdgcn_wmma_f32_16x16x4_f32(false, afrag[kk], false, b0,
                                                   (short)0, c0, false, false);
      }
      epilogue(nt, c0);
    }
  }

  // Wave reduction then one f64 global atomic per wave.
#pragma unroll
  for (int off = 16; off > 0; off >>= 1)
    partial += __shfl_down(partial, off, 32);
  if (lane == 0) atomicAdd(acc_out, partial);
}

// ---------------------------------------------------------------------------
// Kernel 3: scalar finalize
// ---------------------------------------------------------------------------
__global__ void finalize_kernel(const double* __restrict__ acc,
                                const float* __restrict__ sum_vm,
                                float* __restrict__ out) {
  if (threadIdx.x == 0 && blockIdx.x == 0) {
    double norm = (double)(*sum_vm) * (double)kM;
    out[0] = (float)(*acc / norm);
  }
}

extern "C" void kernel_launch(void* const* d_in, const int* in_sizes, int n_in,
                              void* d_out, int out_size, void* d_ws,
                              size_t ws_size, hipStream_t stream) {
  (void)in_sizes; (void)n_in; (void)out_size; (void)ws_size;
  const float* desc  = (const float*)d_in[0];
  const float* wdesc = (const float*)d_in[1];
  const float* homog = (const float*)d_in[2];
  const float* vmask = (const float*)d_in[3];

  char* ws = (char*)d_ws;
  double* acc    = (double*)ws;                                  // 8 B
  float*  sum_vm = (float*)(ws + 8);                             // 4 B (+pad)
  float2* wcc    = (float2*)(ws + 16);                           // B*M*8 B
  float4* ep     = (float4*)(ws + 16 + (size_t)kB * kM * sizeof(float2));

  hipMemsetAsync(d_ws, 0, 16, stream);  // zero accumulators each call

  int total = kB * kM;
  prep_kernel<<<(total + 255) / 256, 256, 0, stream>>>(homog, vmask, wcc, ep,
                                                       sum_vm);

  dim3 grid((kMT + 3) / 4, 8, kB);  // 57 m-tile groups x 8 n-strips x 4 batch
  gemm_loss_kernel<<<grid, 128, 0, stream>>>(desc, wdesc, wcc, ep, acc);

  finalize_kernel<<<1, 1, 0, stream>>>(acc, sum_vm, (float*)d_out);
}